// PointWarping_71863392797315
// MI455X (gfx1250) — compile-verified
//
#include <hip/hip_runtime.h>
#include <math.h>

typedef __attribute__((ext_vector_type(2))) float v2f;
typedef __attribute__((ext_vector_type(8))) float v8f;

#define PW_B 4
#define PW_N 8192

// ---------------------------------------------------------------------------
// Branchless sorted top-3 insert (d0 <= d1 <= d2).
// 3 v_cmp + 10 v_cndmask, no EXEC manipulation -> dual-issue friendly.
// ---------------------------------------------------------------------------
__device__ __forceinline__ void top3_insert(float s, int m,
                                            float& d0, float& d1, float& d2,
                                            int& i0, int& i1, int& i2) {
  const bool lt0 = s < d0;
  const bool lt1 = s < d1;
  const bool lt2 = s < d2;
  const float nd0 = lt0 ? s  : d0;
  const float nd1 = lt0 ? d0 : (lt1 ? s : d1);
  const float nd2 = lt1 ? d1 : (lt2 ? s : d2);
  const int   ni0 = lt0 ? m  : i0;
  const int   ni1 = lt0 ? i0 : (lt1 ? m : i1);
  const int   ni2 = lt1 ? i1 : (lt2 ? m : i2);
  d0 = nd0; d1 = nd1; d2 = nd2;
  i0 = ni0; i1 = ni1; i2 = ni2;
}

// ---------------------------------------------------------------------------
// Kernel 1: pack points for the WMMA distance GEMM.
//   pts[i] = (px, py, pz, |p|^2)      p = xyz1 + flow1   (A-matrix rows)
//   qry[i] = (-2qx, -2qy, -2qz, 1)    q = xyz2           (B-matrix cols)
//   flw[i] = (fx, fy, fz, 0)
// ---------------------------------------------------------------------------
__global__ __launch_bounds__(256) void pw_prep(const float* __restrict__ xyz1,
                                               const float* __restrict__ xyz2,
                                               const float* __restrict__ flow1,
                                               float4* __restrict__ pts,
                                               float4* __restrict__ qry,
                                               float4* __restrict__ flw) {
  int i = blockIdx.x * blockDim.x + threadIdx.x;
  if (i >= PW_B * PW_N) return;
  int b = i / PW_N;
  int n = i - b * PW_N;
  size_t base = (size_t)b * 3 * PW_N + n;

  float fx = flow1[base], fy = flow1[base + PW_N], fz = flow1[base + 2 * PW_N];
  float px = xyz1[base] + fx;
  float py = xyz1[base + PW_N] + fy;
  float pz = xyz1[base + 2 * PW_N] + fz;
  float qx = xyz2[base], qy = xyz2[base + PW_N], qz = xyz2[base + 2 * PW_N];

  pts[i] = make_float4(px, py, pz, px * px + py * py + pz * pz);
  qry[i] = make_float4(-2.0f * qx, -2.0f * qy, -2.0f * qz, 1.0f);
  flw[i] = make_float4(fx, fy, fz, 0.0f);
}

// ---------------------------------------------------------------------------
// Kernel 2: one wave32 per tile of 16 queries.
// D = A(16x4 src) x B(4x16 qry) via V_WMMA_F32_16X16X4_F32 gives
// s(m,j) = |p_m|^2 - 2 p_m.q_j  (argmin-equivalent to dist^2).
// C/D layout: lane l holds, in VGPR v, (M = v + (l<16?0:8), N = l&15)
// -> each lane owns 8 source candidates per tile for one query; the lane
// pair (l, l^16) covers all 16 sources. Per-lane branchless top-3, single
// shfl_xor merge at the end.
// Source loop is software-pipelined one iteration deep: the next pair of
// 16-point tiles is prefetched (unconditionally, with a power-of-two wrap)
// before the current pair's WMMA + ~220-instruction select stream, so the
// s_wait_loadcnt lands after a full iteration of independent VALU work.
// ---------------------------------------------------------------------------
__global__ __launch_bounds__(256) void pw_knn(const float4* __restrict__ pts,
                                              const float4* __restrict__ qry,
                                              const float4* __restrict__ flw,
                                              float* __restrict__ out) {
  const int lane = threadIdx.x & 31;
  const int wave = threadIdx.x >> 5;
  const int tile = blockIdx.x * 8 + wave;     // exact: grid*8 == B*N/16
  const int b    = tile / (PW_N / 16);
  const int j0   = (tile % (PW_N / 16)) * 16;
  const bool lo  = lane < 16;
  const int  ln  = lane & 15;

  const float4* __restrict__ P = pts + (size_t)b * PW_N;
  const float4* __restrict__ Q = qry + (size_t)b * PW_N;
  const float4* __restrict__ F = flw + (size_t)b * PW_N;

  // B operand (4x16, KxN): v0 holds K=0 (lanes 0-15) / K=2 (lanes 16-31),
  //                        v1 holds K=1 (lanes 0-15) / K=3 (lanes 16-31).
  const float4 qv = Q[j0 + ln];
  v2f bmat;
  bmat.x = lo ? qv.x : qv.z;   // -2qx | -2qz
  bmat.y = lo ? qv.y : qv.w;   // -2qy | 1.0

  float d0 = __builtin_inff(), d1 = __builtin_inff(), d2 = __builtin_inff();
  int   i0 = 0, i1 = 0, i2 = 0;
  const int mbase = lo ? 0 : 8;

  // Software pipeline: preload first tile pair.
  float4 pva = P[ln];
  float4 pvb = P[16 + ln];

  for (int m0 = 0; m0 < PW_N; m0 += 32) {
    const float4 cura = pva;
    const float4 curb = pvb;

    // Unconditional prefetch of the next tile pair (wraps to 0 on the last
    // iteration -> one redundant cached load, zero branches).
    const int nx = (m0 + 32) & (PW_N - 1);
    pva = P[nx + ln];
    pvb = P[nx + 16 + ln];

    // A operand (16x4, MxK): v0 = K0|K2, v1 = K1|K3 per half-wave.
    v2f amat_a, amat_b;
    amat_a.x = lo ? cura.x : cura.z;   // px | pz
    amat_a.y = lo ? cura.y : cura.w;   // py | |p|^2
    amat_b.x = lo ? curb.x : curb.z;
    amat_b.y = lo ? curb.y : curb.w;

    v8f ca = {};
    ca = __builtin_amdgcn_wmma_f32_16x16x4_f32(
        false, amat_a, false, bmat, (short)0, ca, false, false);
    v8f cb = {};
    cb = __builtin_amdgcn_wmma_f32_16x16x4_f32(
        false, amat_b, false, bmat, (short)0, cb, false, false);

#pragma unroll
    for (int v = 0; v < 8; ++v)
      top3_insert(ca[v], m0 + mbase + v, d0, d1, d2, i0, i1, i2);
#pragma unroll
    for (int v = 0; v < 8; ++v)
      top3_insert(cb[v], m0 + 16 + mbase + v, d0, d1, d2, i0, i1, i2);
  }

  // Merge the two half-wave top-3 lists (lane l <-> lane l^16).
  float pd0 = __shfl_xor(d0, 16, 32); int pi0 = __shfl_xor(i0, 16, 32);
  float pd1 = __shfl_xor(d1, 16, 32); int pi1 = __shfl_xor(i1, 16, 32);
  float pd2 = __shfl_xor(d2, 16, 32); int pi2 = __shfl_xor(i2, 16, 32);
  top3_insert(pd0, pi0, d0, d1, d2, i0, i1, i2);
  top3_insert(pd1, pi1, d0, d1, d2, i0, i1, i2);
  top3_insert(pd2, pi2, d0, d1, d2, i0, i1, i2);

  // Lanes 0-15 finish their query: exact IDW interpolation + output.
  if (lo) {
    const int j = j0 + ln;
    const float qx = -0.5f * qv.x, qy = -0.5f * qv.y, qz = -0.5f * qv.z;
    const int idx[3] = {i0, i1, i2};
    float w[3], wsum = 0.0f;
#pragma unroll
    for (int k = 0; k < 3; ++k) {
      const float4 p = P[idx[k]];
      const float dx = p.x - qx, dy = p.y - qy, dz = p.z - qz;
      float dist = sqrtf(dx * dx + dy * dy + dz * dz);
      dist = fmaxf(dist, 1e-10f);
      w[k] = 1.0f / dist;
      wsum += w[k];
    }
    const float inv = 1.0f / wsum;
    float fx = 0.0f, fy = 0.0f, fz = 0.0f;
#pragma unroll
    for (int k = 0; k < 3; ++k) {
      const float4 f = F[idx[k]];
      const float ww = w[k] * inv;
      fx += ww * f.x; fy += ww * f.y; fz += ww * f.z;
    }
    const size_t ob = (size_t)b * 3 * PW_N + j;
    out[ob]            = qx - fx;
    out[ob + PW_N]     = qy - fy;
    out[ob + 2 * PW_N] = qz - fz;
  }
}

// ---------------------------------------------------------------------------
extern "C" void kernel_launch(void* const* d_in, const int* in_sizes, int n_in,
                              void* d_out, int out_size, void* d_ws, size_t ws_size,
                              hipStream_t stream) {
  const float* xyz1  = (const float*)d_in[0];
  const float* xyz2  = (const float*)d_in[1];
  const float* flow1 = (const float*)d_in[2];
  float* out = (float*)d_out;

  float4* pts = (float4*)d_ws;             // B*N float4
  float4* qry = pts + PW_B * PW_N;         // B*N float4
  float4* flw = qry + PW_B * PW_N;         // B*N float4  (total 1.5 MB)

  pw_prep<<<(PW_B * PW_N + 255) / 256, 256, 0, stream>>>(xyz1, xyz2, flow1,
                                                         pts, qry, flw);
  // B*N/16 = 2048 query tiles, 8 waves (one tile each) per 256-thread block.
  pw_knn<<<(PW_B * PW_N / 16) / 8, 256, 0, stream>>>(pts, qry, flw, out);
}